// RuleFocusedAttention_74191265071677
// MI455X (gfx1250) — compile-verified
//
#include <hip/hip_runtime.h>
#include <cmath>

#define HIDDEN 128
#define LDS_STRIDE 130   // pad rows to avoid LDS bank conflicts (64 banks x 4B)

typedef __attribute__((ext_vector_type(2))) float v2f;
typedef __attribute__((ext_vector_type(8))) float v8f;

// Workspace layout (floats):
//  [0..127]   sum over light rows
//  [128..255] sum over stop rows
//  [256]      count light
//  [257]      count stop
//  [320..447] cvec = b1 + W1b @ h_light + W1c @ h_stop  (folded constant)

__global__ void k_init(float* __restrict__ ws) {
    for (int i = threadIdx.x; i < 576; i += blockDim.x) ws[i] = 0.0f;
}

// Block = 256 threads laid out as 2 row-groups x 128 columns; each block owns 128 rows.
__global__ __launch_bounds__(256) void k_reduce(const float* __restrict__ h,
                                                const int* __restrict__ et,
                                                int N, float* __restrict__ ws) {
    const int col = threadIdx.x & 127;
    const int rg  = threadIdx.x >> 7;
    const int rowBase = blockIdx.x * 128;
    float sl = 0.f, ss = 0.f, cl = 0.f, cs = 0.f;
    for (int i = rg; i < 128; i += 2) {
        const int row = rowBase + i;
        if (row < N) {
            const int t = et[row];
            const float v = h[(size_t)row * HIDDEN + col];
            if (t == 1) { sl += v; if (col == 0) cl += 1.f; }
            else if (t == 2) { ss += v; if (col == 0) cs += 1.f; }
        }
    }
    __shared__ float red[512];
    red[threadIdx.x]       = sl;
    red[256 + threadIdx.x] = ss;
    __syncthreads();
    if (rg == 0) {
        sl += red[threadIdx.x + 128];
        ss += red[256 + threadIdx.x + 128];
        atomicAdd(&ws[col], sl);
        atomicAdd(&ws[HIDDEN + col], ss);
    }
    if (col == 0) {
        atomicAdd(&ws[256], cl);
        atomicAdd(&ws[257], cs);
    }
}

// One block, 128 threads: fold the broadcast-term of the first linear layer.
__global__ void k_prep(const float* __restrict__ W1, const float* __restrict__ b1,
                       float* __restrict__ ws) {
    __shared__ float hl[HIDDEN], hs[HIDDEN];
    const int t = threadIdx.x;
    const float cl = ws[256], cs = ws[257];
    hl[t] = cl > 0.f ? ws[t]          / fmaxf(cl, 1.f) : 0.f;
    hs[t] = cs > 0.f ? ws[HIDDEN + t] / fmaxf(cs, 1.f) : 0.f;
    __syncthreads();
    const float* wrow = W1 + (size_t)t * (3 * HIDDEN);
    float acc = b1[t];
    #pragma unroll 4
    for (int k = 0; k < HIDDEN; ++k)
        acc += wrow[HIDDEN + k] * hl[k] + wrow[2 * HIDDEN + k] * hs[k];
    ws[320 + t] = acc;
}

// Main fused kernel: 256 threads = 8 waves; each wave owns a 16-row tile; block owns 128 rows.
// hid[16x128] = h_tile[16x128] @ W1a^T via 32 K-steps x 8 N-tiles of v_wmma_f32_16x16x4_f32,
// then relu, dot W2, butterfly-reduce, sigmoid, blend, store.
__global__ __launch_bounds__(256) void k_main(
    const float* __restrict__ h, const int* __restrict__ et,
    const int* __restrict__ rule_id_p,
    const float* __restrict__ W1, const float* __restrict__ W2,
    const float* __restrict__ b2, const float* __restrict__ rule_emb,
    const float* __restrict__ ws,
    float* __restrict__ out_h, float* __restrict__ out_beta, int N) {

    extern __shared__ float smem[];
    float* ldsW = smem;                          // [128][130] W1a, n-major
    float* ldsH = smem + HIDDEN * LDS_STRIDE;    // [128][130] h tile, row-major

    const int tid = threadIdx.x;
    const int rowBase = blockIdx.x * 128;

    // Stage W1a (cols 0..127 of W1, row stride 384) and the 128x128 h tile.
    for (int idx = tid; idx < HIDDEN * (HIDDEN / 4); idx += 256) {
        const int n  = idx >> 5;
        const int k4 = (idx & 31) << 2;
        const float4 w = *(const float4*)(W1 + (size_t)n * (3 * HIDDEN) + k4);
        float* dst = ldsW + n * LDS_STRIDE + k4;
        dst[0] = w.x; dst[1] = w.y; dst[2] = w.z; dst[3] = w.w;
    }
    for (int idx = tid; idx < HIDDEN * (HIDDEN / 4); idx += 256) {
        const int r  = idx >> 5;
        const int k4 = (idx & 31) << 2;
        int row = rowBase + r; if (row >= N) row = N - 1;   // clamp, keep EXEC full
        const float4 v = *(const float4*)(h + (size_t)row * HIDDEN + k4);
        float* dst = ldsH + r * LDS_STRIDE + k4;
        dst[0] = v.x; dst[1] = v.y; dst[2] = v.z; dst[3] = v.w;
    }
    __syncthreads();

    const int wave = tid >> 5;
    const int lane = tid & 31;
    const int l15  = lane & 15;
    const int half = lane >> 4;               // K sub-offset selector (A/B layout)
    const float* hrow = ldsH + (wave * 16 + l15) * LDS_STRIDE;

    v8f acc[8] = {};   // 8 N-tiles of the 16x128 hid tile, fp32 accumulators

    for (int kt = 0; kt < 32; ++kt) {
        const int k = kt * 4 + half * 2;
        const v2f a = *(const v2f*)(hrow + k);     // A: M=l15, K={k,k+1}
        #pragma unroll
        for (int nt = 0; nt < 8; ++nt) {
            const v2f b = *(const v2f*)(ldsW + (nt * 16 + l15) * LDS_STRIDE + k); // B = W1a^T frag
            acc[nt] = __builtin_amdgcn_wmma_f32_16x16x4_f32(
                false, a, false, b, (short)0, acc[nt], false, false);
        }
    }

    // Epilogue: hid = relu(acc + cvec[n]); p[r] = sum_n W2[n]*hid  (per-lane partial over 8 cols)
    float p[8];
    #pragma unroll
    for (int r = 0; r < 8; ++r) p[r] = 0.f;
    #pragma unroll
    for (int nt = 0; nt < 8; ++nt) {
        const int n = nt * 16 + l15;
        const float cn  = ws[320 + n];
        const float w2n = W2[n];
        #pragma unroll
        for (int r = 0; r < 8; ++r)
            p[r] += w2n * fmaxf(acc[nt][r] + cn, 0.f);
    }
    // Reduce across the 16 lanes of each half (rows 0..7 live in lanes 0-15, rows 8..15 in 16-31)
    #pragma unroll
    for (int m = 1; m < 16; m <<= 1) {
        #pragma unroll
        for (int r = 0; r < 8; ++r) p[r] += __shfl_xor(p[r], m, 32);
    }
    const float b2v = b2[0];
    float beta[8];
    #pragma unroll
    for (int r = 0; r < 8; ++r)
        beta[r] = 1.0f / (1.0f + __expf(-(p[r] + b2v)));

    // Outputs: each lane stores float4 (cols 4*lane..4*lane+3) per row.
    const int ruleId = rule_id_p[0];
    const float4 er = *(const float4*)(rule_emb + (size_t)ruleId * HIDDEN + lane * 4);
    #pragma unroll
    for (int m = 0; m < 16; ++m) {
        const float bm = __shfl(beta[m & 7], (m >> 3) << 4, 32);
        const int row = rowBase + wave * 16 + m;
        if (row < N) {
            const bool car = (et[row] == 0);
            const float* hr = ldsH + (wave * 16 + m) * LDS_STRIDE + lane * 4;
            float4 hv; hv.x = hr[0]; hv.y = hr[1]; hv.z = hr[2]; hv.w = hr[3];
            float4 o;
            if (car) {
                const float ob = 1.0f - bm;
                o.x = bm * hv.x + ob * er.x;
                o.y = bm * hv.y + ob * er.y;
                o.z = bm * hv.z + ob * er.z;
                o.w = bm * hv.w + ob * er.w;
            } else {
                o = hv;
            }
            *(float4*)(out_h + (size_t)row * HIDDEN + lane * 4) = o;
            if (lane == 0) out_beta[row] = car ? bm : 0.f;
        }
    }
}

extern "C" void kernel_launch(void* const* d_in, const int* in_sizes, int n_in,
                              void* d_out, int out_size, void* d_ws, size_t ws_size,
                              hipStream_t stream) {
    const float* h       = (const float*)d_in[0];
    const int*   et      = (const int*)  d_in[1];
    const int*   rule_id = (const int*)  d_in[2];
    const float* W1      = (const float*)d_in[3];
    const float* b1      = (const float*)d_in[4];
    const float* W2      = (const float*)d_in[5];
    const float* b2      = (const float*)d_in[6];
    const float* remb    = (const float*)d_in[7];

    const int N = in_sizes[0] / HIDDEN;
    float* ws       = (float*)d_ws;
    float* out_h    = (float*)d_out;
    float* out_beta = out_h + (size_t)N * HIDDEN;

    const int nblk = (N + 127) / 128;
    const size_t lds_bytes = (size_t)2 * HIDDEN * LDS_STRIDE * sizeof(float); // 130 KB

    k_init  <<<1, 256, 0, stream>>>(ws);
    k_reduce<<<nblk, 256, 0, stream>>>(h, et, N, ws);
    k_prep  <<<1, 128, 0, stream>>>(W1, b1, ws);
    k_main  <<<nblk, 256, lds_bytes, stream>>>(h, et, rule_id, W1, W2, b2, remb, ws,
                                               out_h, out_beta, N);
}